// QLayerOriginal_40578851013035
// MI455X (gfx1250) — compile-verified
//
#include <hip/hip_runtime.h>
#include <hip/hip_bf16.h>

#define THREADS 256
#define DFEAT   48     // D == D3 == 48
#define NBLK    16     // 16 blocks of 3 angles
#define ROWPAD  52     // 48 -> 52 floats/row: keeps 16B alignment for b128 LDS
                       // writes, reduces read bank conflicts to 2-way

// v_sin_f32 / v_cos_f32 take the angle pre-scaled by 1/(2*pi)
__device__ __forceinline__ void fsincos(float x, float& s, float& c) {
    const float t = x * 0.15915494309189535f;
    s = __builtin_amdgcn_sinf(t);
    c = __builtin_amdgcn_cosf(t);
}

__global__ __launch_bounds__(THREADS, 1) void qlayer_su2_kernel(
    const float* __restrict__ rho_re,
    const float* __restrict__ rho_im,
    const float* __restrict__ x,
    const float* __restrict__ w,
    const float* __restrict__ theta,
    float* __restrict__ out,
    int Btot)
{
    __shared__ __align__(16) float xs[THREADS * ROWPAD];

    const int tid = threadIdx.x;
    const long long b = (long long)blockIdx.x * THREADS + tid;

    // ---------------------------------------------------------------
    // Stage this block's x tile (256 rows x 48 floats = 48KB) into LDS
    // via the CDNA5 async DMA path (ASYNCcnt). Global reads are fully
    // coalesced 16B/lane; LDS destination rows padded to 52 floats.
    // ---------------------------------------------------------------
    const long long tile_begin = (long long)blockIdx.x * (THREADS * DFEAT);
    const long long total_x    = (long long)Btot * DFEAT;
    const bool full_tile = (tile_begin + (long long)(THREADS * DFEAT)) <= total_x;

    if (full_tile) {
        const float* gx = x + tile_begin;                    // uniform -> SGPR pair
        const unsigned lds_base = (unsigned)(size_t)(&xs[0]); // low 32b of flat shared ptr == LDS byte offset
        // chunk q = i*256 + tid covers global floats [4q, 4q+4);
        // row r = q/12 (12 x 16B chunks per 48-float row), col c = q%12.
        int r = tid / 12;
        int c = tid - r * 12;
        const unsigned voff0 = (unsigned)tid * 16u;
        #pragma unroll
        for (int i = 0; i < 12; ++i) {
            const unsigned ldsb  = lds_base + (unsigned)(r * (ROWPAD * 4) + c * 16);
            const unsigned voffi = voff0 + (unsigned)(i * 4096);
            asm volatile("global_load_async_to_lds_b128 %0, %1, %2 offset:0"
                         :: "v"(ldsb), "v"(voffi), "s"(gx)
                         : "memory");
            // advance q by 256 chunks: 256 = 21*12 + 4
            r += 21; c += 4;
            if (c >= 12) { c -= 12; r += 1; }
        }
    } else {
        // ragged tail block (not hit for B = 1M, kept for safety)
        for (int idx = tid; idx < THREADS * DFEAT; idx += THREADS) {
            const int rl = idx / DFEAT;
            const int j  = idx - rl * DFEAT;
            const long long g = tile_begin + idx;
            xs[rl * ROWPAD + j] = (g < total_x) ? x[g] : 0.0f;
        }
    }

    // Overlap: issue the perfectly-coalesced rho loads while the async
    // engine fills LDS.
    float4 rr = make_float4(0.f, 0.f, 0.f, 0.f);
    float4 ri = make_float4(0.f, 0.f, 0.f, 0.f);
    const bool active = (b < (long long)Btot);
    if (active) {
        rr = reinterpret_cast<const float4*>(rho_re)[b];  // r00 r01 r10 r11
        ri = reinterpret_cast<const float4*>(rho_im)[b];
    }

    if (full_tile) {
        asm volatile("s_wait_asynccnt 0x0" ::: "memory");
    }
    __syncthreads();

    if (active) {
        // ---------------------------------------------------------------
        // Compose V = U16 * ... * U1 in SU(2) form V = [[A,-B],[B*,A*]]:
        //   A' = alpha*A - beta*conj(B);  B' = alpha*B + beta*conj(A)
        // with alpha = cos(a) e^{ib}, beta = sin(a) e^{ic}.
        // ---------------------------------------------------------------
        const float* xrow = &xs[tid * ROWPAD];
        float Ar = 1.0f, Ai = 0.0f, Br = 0.0f, Bi = 0.0f;

        #pragma unroll
        for (int k = 0; k < NBLK; ++k) {
            const int j = 3 * k;
            const float p0 = fmaf(w[j + 0], xrow[j + 0], theta[j + 0]);
            const float p1 = fmaf(w[j + 1], xrow[j + 1], theta[j + 1]);
            const float p2 = fmaf(w[j + 2], xrow[j + 2], theta[j + 2]);

            float sa, ca, s1, c1, s2, c2;
            fsincos(0.5f * p0, sa, ca);      // a = p0/2
            fsincos(0.5f * p1, s1, c1);      // h1 = p1/2
            fsincos(0.5f * p2, s2, c2);      // h2 = p2/2
            // b = h1+h2, c = h1-h2 via angle-sum (saves 2 sincos pairs)
            const float cb = c1 * c2 - s1 * s2, sb = s1 * c2 + c1 * s2;
            const float cc = c1 * c2 + s1 * s2, sc = s1 * c2 - c1 * s2;

            const float ar = ca * cb, ai = ca * sb;   // alpha
            const float br = sa * cc, bi = sa * sc;   // beta

            const float nAr = ar * Ar - ai * Ai - br * Br - bi * Bi;
            const float nAi = ar * Ai + ai * Ar - bi * Br + br * Bi;
            const float nBr = ar * Br - ai * Bi + br * Ar + bi * Ai;
            const float nBi = ar * Bi + ai * Br + bi * Ar - br * Ai;
            Ar = nAr; Ai = nAi; Br = nBr; Bi = nBi;
        }

        // ---------------- rho_out = V * rho * V^dagger -----------------
        // M = V * rho,  V = [[A,-B],[B*,A*]]
        const float m00r = (Ar * rr.x - Ai * ri.x) - (Br * rr.z - Bi * ri.z);
        const float m00i = (Ar * ri.x + Ai * rr.x) - (Br * ri.z + Bi * rr.z);
        const float m01r = (Ar * rr.y - Ai * ri.y) - (Br * rr.w - Bi * ri.w);
        const float m01i = (Ar * ri.y + Ai * rr.y) - (Br * ri.w + Bi * rr.w);
        const float m10r = (Br * rr.x + Bi * ri.x) + (Ar * rr.z + Ai * ri.z);
        const float m10i = (Br * ri.x - Bi * rr.x) + (Ar * ri.z - Ai * rr.z);
        const float m11r = (Br * rr.y + Bi * ri.y) + (Ar * rr.w + Ai * ri.w);
        const float m11i = (Br * ri.y - Bi * rr.y) + (Ar * ri.w - Ai * rr.w);

        // out = M * V^dagger,  V^dagger = [[A*, B],[-B*, A]]
        const float o00r = (m00r * Ar + m00i * Ai) - (m01r * Br + m01i * Bi);
        const float o00i = (m00i * Ar - m00r * Ai) - (m01i * Br - m01r * Bi);
        const float o01r = (m00r * Br - m00i * Bi) + (m01r * Ar - m01i * Ai);
        const float o01i = (m00r * Bi + m00i * Br) + (m01r * Ai + m01i * Ar);
        const float o10r = (m10r * Ar + m10i * Ai) - (m11r * Br + m11i * Bi);
        const float o10i = (m10i * Ar - m10r * Ai) - (m11i * Br - m11r * Bi);
        const float o11r = (m10r * Br - m10i * Bi) + (m11r * Ar - m11i * Ai);
        const float o11i = (m10r * Bi + m10i * Br) + (m11r * Ai + m11i * Ar);

        // complex64 [B,2,2] -> interleaved re/im, 8 floats per element
        reinterpret_cast<float4*>(out)[2 * b + 0] = make_float4(o00r, o00i, o01r, o01i);
        reinterpret_cast<float4*>(out)[2 * b + 1] = make_float4(o10r, o10i, o11r, o11i);
    }
}

extern "C" void kernel_launch(void* const* d_in, const int* in_sizes, int n_in,
                              void* d_out, int out_size, void* d_ws, size_t ws_size,
                              hipStream_t stream) {
    const float* rho_re = (const float*)d_in[0];  // [B,2,2]
    const float* rho_im = (const float*)d_in[1];  // [B,2,2]
    const float* x      = (const float*)d_in[2];  // [B,48]
    const float* w      = (const float*)d_in[3];  // [48]
    const float* theta  = (const float*)d_in[4];  // [48]
    float* out = (float*)d_out;                   // [B,2,2] complex64 -> 8 floats/elt

    const int B = in_sizes[0] / 4;
    const int grid = (B + THREADS - 1) / THREADS;
    qlayer_su2_kernel<<<grid, THREADS, 0, stream>>>(rho_re, rho_im, x, w, theta, out, B);
}